// GatedDeltaMixer_56985626084080
// MI455X (gfx1250) — compile-verified
//
#include <hip/hip_runtime.h>
#include <cstdint>
#include <cstddef>

// ---------------------------------------------------------------------------
// GatedDeltaMixer for MI455X (gfx1250), fp32 end-to-end.
// GEMMs: V_WMMA_F32_16X16X4_F32 (native fp32 matrix op, wave32) with
// double-buffered LDS staged via GLOBAL_LOAD_ASYNC_TO_LDS_B128 when available.
// Problem sizes (fixed by reference): B=2 S=4096 D=2048 H=16 hd=128 C=64.
// ---------------------------------------------------------------------------

typedef __attribute__((ext_vector_type(2))) float v2f;
typedef __attribute__((ext_vector_type(8))) float v8f;

#ifndef __has_builtin
#define __has_builtin(x) 0
#endif
#if __has_builtin(__builtin_amdgcn_global_load_async_to_lds_b128)
#define USE_ASYNC_LDS 1
#else
#define USE_ASYNC_LDS 0
#endif

#define BB   2
#define SS   4096
#define DD   2048
#define HH   16
#define HD   128
#define CK   64            // chunk length
#define NCH  (SS / CK)     // 64 chunks
#define RMS_EPS 1.1920928955078125e-07f

// ---- GEMM tiling ----------------------------------------------------------
#define TM 128             // workgroup tile M
#define TN 256             // workgroup tile N
#define KB 16              // k-block staged in LDS
#define LDP (KB + 4)       // LDS row pitch (floats): stride 20 -> 16 distinct
                           // banks across the 16 fragment rows (gcd(20,64)=4)

__device__ __forceinline__ float sigmoidf_(float x) {
    return 1.0f / (1.0f + __expf(-x));
}

#if USE_ASYNC_LDS
// Builtin expects: (int4-vector __device__(AS1) *, int4-vector __shared__(AS3) *,
//                   imm offset, imm cpol) — per hipcc diagnostic round 2.
typedef int v4i_vs __attribute__((vector_size(16)));
typedef __attribute__((address_space(1))) v4i_vs g_v4i;
typedef __attribute__((address_space(3))) v4i_vs l_v4i;

// One 16-byte async copy, ASYNCcnt-tracked, VGPR-free.
// Generic LDS pointer low 32 bits == LDS byte offset (ISA aperture rule);
// generic global pointer == global address.
__device__ __forceinline__ void async_copy16(void* lds, const void* gmem) {
    __builtin_amdgcn_global_load_async_to_lds_b128(
        (g_v4i*)(uintptr_t)gmem,
        (l_v4i*)(uint32_t)(uintptr_t)lds,
        /*offset=*/0, /*cpol=*/0);
}
#endif

__device__ __forceinline__ void wait_stage() {
#if USE_ASYNC_LDS
#if __has_builtin(__builtin_amdgcn_s_wait_asynccnt)
    __builtin_amdgcn_s_wait_asynccnt(0);
#else
    asm volatile("s_wait_asynccnt 0" ::: "memory");
#endif
#endif
}

// Stage one 128x16 A tile + 256x16 B tile into an LDS buffer.
__device__ __forceinline__ void stage_tile(const float* __restrict__ A,
                                           const float* __restrict__ Bm,
                                           float (*As)[LDP], float (*Bs)[LDP],
                                           int tid, int mBase, int nBase,
                                           int K, int k0) {
    const int row = tid >> 2;            // 0..63
    const int col = (tid & 3) * 4;       // 0,4,8,12
#if USE_ASYNC_LDS
    async_copy16(&As[row][col],      &A[(size_t)(mBase + row) * K + k0 + col]);
    async_copy16(&As[row + 64][col], &A[(size_t)(mBase + row + 64) * K + k0 + col]);
#pragma unroll
    for (int i = 0; i < 4; ++i) {
        const int r = row + 64 * i;
        async_copy16(&Bs[r][col], &Bm[(size_t)(nBase + r) * K + k0 + col]);
    }
#else
    *(float4*)(&As[row][col]) =
        *(const float4*)(&A[(size_t)(mBase + row) * K + k0 + col]);
    *(float4*)(&As[row + 64][col]) =
        *(const float4*)(&A[(size_t)(mBase + row + 64) * K + k0 + col]);
#pragma unroll
    for (int i = 0; i < 4; ++i) {
        const int r = row + 64 * i;
        *(float4*)(&Bs[r][col]) =
            *(const float4*)(&Bm[(size_t)(nBase + r) * K + k0 + col]);
    }
#endif
}

// C[M,N] = A[M,K] * B[N,K]^T   (both row-major, "NT" gemm: K contiguous both)
__global__ __launch_bounds__(256)
void gemm_nt_f32_wmma(const float* __restrict__ A, const float* __restrict__ Bm,
                      float* __restrict__ C, int M, int N, int K)
{
    __shared__ float As[2][TM][LDP];   // 2 x 10 KB
    __shared__ float Bs[2][TN][LDP];   // 2 x 20 KB

    const int tid  = threadIdx.x;
    const int lane = tid & 31;
    const int wave = tid >> 5;          // 8 waves
    const int wm   = (wave & 1) * 64;   // wave tile origin in workgroup tile
    const int wn   = (wave >> 1) * 64;
    const int mBase = blockIdx.y * TM;
    const int nBase = blockIdx.x * TN;

    const int lr   = lane & 15;         // row-in-fragment / N-in-fragment
    const int lh   = lane >> 4;         // half-wave select
    const int colk = lh * 2;            // K sub-offset per ISA 16x4 f32 A layout

    v8f acc[4][4] = {};                 // 4x4 fragments of 16x16 = 64x64/wave

    // Prologue: stage k-block 0 into buffer 0.
    stage_tile(A, Bm, As[0], Bs[0], tid, mBase, nBase, K, 0);

    int cur = 0;
    for (int k0 = 0; k0 < K; k0 += KB) {
        wait_stage();          // my async copies into As[cur]/Bs[cur] done
        __syncthreads();       // everyone's copies visible

        if (k0 + KB < K) {
            // Overlap: stage next k-block into the other buffer while
            // this buffer feeds the WMMA pipe.
            stage_tile(A, Bm, As[cur ^ 1], Bs[cur ^ 1], tid,
                       mBase, nBase, K, k0 + KB);
            // Warm far cache two blocks ahead (global_prefetch_b8).
            if (k0 + 2 * KB < K) {
                __builtin_prefetch(&A[(size_t)(mBase + (tid >> 2)) * K + k0 + 2 * KB], 0, 0);
                __builtin_prefetch(&Bm[(size_t)(nBase + (tid >> 2)) * K + k0 + 2 * KB], 0, 0);
            }
        }

        // ---- 4 WMMA k-steps of depth 4 each on the current buffer ----
        const float (*Ac)[LDP] = As[cur];
        const float (*Bc)[LDP] = Bs[cur];
#pragma unroll
        for (int kk = 0; kk < KB / 4; ++kk) {
            v2f afrag[4], bfrag[4];
#pragma unroll
            for (int i = 0; i < 4; ++i) {
                // A 16x4 f32 layout: lane<16 -> K=0,1 ; lane>=16 -> K=2,3
                const float* p = &Ac[wm + i * 16 + lr][kk * 4 + colk];
                v2f a; a.x = p[0]; a.y = p[1];
                afrag[i] = a;
            }
#pragma unroll
            for (int j = 0; j < 4; ++j) {
                // B 4x16 f32 layout mirrors A (row striped across lanes)
                const float* p = &Bc[wn + j * 16 + lr][kk * 4 + colk];
                v2f b; b.x = p[0]; b.y = p[1];
                bfrag[j] = b;
            }
#pragma unroll
            for (int i = 0; i < 4; ++i)
#pragma unroll
                for (int j = 0; j < 4; ++j)
                    acc[i][j] = __builtin_amdgcn_wmma_f32_16x16x4_f32(
                        /*neg_a=*/false, afrag[i],
                        /*neg_b=*/false, bfrag[j],
                        /*c_mod=*/(short)0, acc[i][j],
                        /*reuse_a=*/false, /*reuse_b=*/false);
        }
        cur ^= 1;
    }

    // ---- epilogue: C/D 16x16 f32: VGPR r -> M = r + 8*(lane>>4), N = lane&15
    const int mlo = 8 * lh;
#pragma unroll
    for (int i = 0; i < 4; ++i) {
#pragma unroll
        for (int j = 0; j < 4; ++j) {
            const int gn = nBase + wn + j * 16 + lr;
#pragma unroll
            for (int r = 0; r < 8; ++r) {
                const int gm = mBase + wm + i * 16 + mlo + r;
                C[(size_t)gm * N + gn] = acc[i][j][r];
            }
        }
    }
}

// ---------------------------------------------------------------------------
// Pass 2: per-(b,h,chunk) chunk summaries.  Faithful to the reference chunked
// scan: prefix = cumprod(a), acc = cumsum(u / max(prefix,1e-6)).
// Block = 128 threads (one per hd channel), grid = B*H*NC.
// proj layout: [(b*S+s), 5*D] with sections q,k,v,a,g at n = sec*D + h*HD + d.
// ---------------------------------------------------------------------------
__global__ __launch_bounds__(128)
void chunk_summary_kernel(const float* __restrict__ proj,
                          float* __restrict__ P, float* __restrict__ Acc)
{
    const int blk = blockIdx.x;              // ((c*B + b)*H + h)
    const int h = blk % HH;
    const int t1 = blk / HH;
    const int b = t1 % BB;
    const int c = t1 / BB;
    const int d = threadIdx.x;

    float prefix = 1.0f, acc = 0.0f;
    for (int t = 0; t < CK; ++t) {
        const int s = c * CK + t;
        const size_t base = (size_t)(b * SS + s) * (5 * DD) + h * HD + d;
        const float kv = proj[base + DD];
        const float vv = proj[base + 2 * DD];
        const float av = proj[base + 3 * DD];
        float a = sigmoidf_(av + 2.0f);
        a = fminf(fmaxf(a, 0.6f), 0.9995f);
        const float u = tanhf(kv) * vv;
        prefix *= a;
        acc += u / fmaxf(prefix, 1e-6f);
    }
    const size_t idx = (size_t)c * (BB * HH * HD) + (size_t)b * (HH * HD) + h * HD + d;
    P[idx] = prefix;
    Acc[idx] = acc;
}

// ---------------------------------------------------------------------------
// Pass 3: sequential scan over NC=64 chunks per channel (4096 channels).
// state_{c+1} = P_c * (state_c + A_c); store chunk-ENTRY state.
// ---------------------------------------------------------------------------
__global__ __launch_bounds__(256)
void chunk_scan_kernel(const float* __restrict__ P, const float* __restrict__ Acc,
                       float* __restrict__ Entry)
{
    const int ch = blockIdx.x * blockDim.x + threadIdx.x;   // 0 .. B*H*HD-1
    float st = 0.0f;
    const int stride = BB * HH * HD;
    for (int c = 0; c < NCH; ++c) {
        const size_t idx = (size_t)c * stride + ch;
        Entry[idx] = st;
        st = P[idx] * (st + Acc[idx]);
    }
}

// ---------------------------------------------------------------------------
// Pass 4: in-chunk replay + RMS-norm(q) + gating -> y[B,S,D].
// RMS over hd=128 = wave32 __shfl_xor butterfly + 4-wave LDS combine.
// ---------------------------------------------------------------------------
__global__ __launch_bounds__(128)
void mixer_final_kernel(const float* __restrict__ proj,
                        const float* __restrict__ Entry, float* __restrict__ Y)
{
    __shared__ float red[4];
    const int blk = blockIdx.x;
    const int h = blk % HH;
    const int t1 = blk / HH;
    const int b = t1 % BB;
    const int c = t1 / BB;
    const int d = threadIdx.x;
    const int lane = d & 31;
    const int wv = d >> 5;

    const int ch = b * (HH * HD) + h * HD + d;
    const float state = Entry[(size_t)c * (BB * HH * HD) + ch];

    float prefix = 1.0f, acc = 0.0f;
    for (int t = 0; t < CK; ++t) {
        const int s = c * CK + t;
        const size_t base = (size_t)(b * SS + s) * (5 * DD) + h * HD + d;
        const float qv = proj[base];
        const float kv = proj[base + DD];
        const float vv = proj[base + 2 * DD];
        const float av = proj[base + 3 * DD];
        const float gv = proj[base + 4 * DD];

        // ---- RMS-norm of q over hd=128 ----
        float sq = qv * qv;
#pragma unroll
        for (int o = 16; o > 0; o >>= 1) sq += __shfl_xor(sq, o, 32);
        if (lane == 0) red[wv] = sq;
        __syncthreads();
        const float tot = red[0] + red[1] + red[2] + red[3];
        __syncthreads();
        const float qn = qv * rsqrtf(tot * (1.0f / HD) + RMS_EPS);

        float a = sigmoidf_(av + 2.0f);
        a = fminf(fmaxf(a, 0.6f), 0.9995f);
        const float u = tanhf(kv) * vv;
        prefix *= a;
        acc += u / fmaxf(prefix, 1e-6f);
        const float mem = prefix * (state + acc);

        const float g = sigmoidf_(gv);
        Y[(size_t)(b * SS + s) * DD + h * HD + d] = g * (qn * mem) + (1.0f - g) * vv;
    }
}

// ---------------------------------------------------------------------------
// Launch: GEMM1 -> summaries -> chunk scan -> final mix -> GEMM2.
// Workspace (floats): proj 8192*10240, y 8192*2048, P/A/Entry 64*4096 each
//   ~406 MB total, carved from d_ws.
// ---------------------------------------------------------------------------
extern "C" void kernel_launch(void* const* d_in, const int* in_sizes, int n_in,
                              void* d_out, int out_size, void* d_ws, size_t ws_size,
                              hipStream_t stream)
{
    (void)in_sizes; (void)n_in; (void)out_size; (void)ws_size;

    const float* x     = (const float*)d_in[0];
    const float* w_in  = (const float*)d_in[1];
    const float* w_out = (const float*)d_in[2];
    float* out = (float*)d_out;

    const size_t M     = (size_t)BB * SS;   // 8192 rows
    const size_t projN = (size_t)5 * DD;    // 10240

    float* ws    = (float*)d_ws;
    float* proj  = ws;                                   // M * projN
    float* y     = proj + M * projN;                     // M * DD
    float* P     = y + M * DD;                           // NCH * B*H*HD
    float* Acc   = P + (size_t)NCH * BB * HH * HD;
    float* Entry = Acc + (size_t)NCH * BB * HH * HD;

    // GEMM1: proj = x @ w_in^T
    dim3 g1(projN / TN, M / TM);             // (40, 64)
    gemm_nt_f32_wmma<<<g1, 256, 0, stream>>>(x, w_in, proj, (int)M, (int)projN, DD);

    // Chunked scan (faithful to reference clamped-cumsum math)
    chunk_summary_kernel<<<BB * HH * NCH, 128, 0, stream>>>(proj, P, Acc);
    chunk_scan_kernel<<<(BB * HH * HD) / 256, 256, 0, stream>>>(P, Acc, Entry);
    mixer_final_kernel<<<BB * HH * NCH, 128, 0, stream>>>(proj, Entry, y);

    // GEMM2: out = y @ w_out^T
    dim3 g2(DD / TN, M / TM);                // (8, 64)
    gemm_nt_f32_wmma<<<g2, 256, 0, stream>>>(y, w_out, out, (int)M, DD, DD);
}